// GNN_45011257262538
// MI455X (gfx1250) — compile-verified
//
#include <hip/hip_runtime.h>
#include <hip/hip_bf16.h>
#include <stdint.h>

// ---------------------------------------------------------------------------
// GIN-style GNN, 5 layers, EMB=128, fp32, gfx1250 (MI455X).
//   h = x@Wx + bx                                  (f32 WMMA GEMM)
//   per layer:
//     agg[dst] += h[src] + (ea@We + be)            (fused gather/scatter, LDS-staged We)
//     h2 = relu(agg@W1 + b1)@W2 + b2               (fused WMMA MLP through LDS)
//     BN stats folded into MLP kernel (atomics), then elementwise BN apply.
// Tile staging uses GLOBAL_LOAD_ASYNC_TO_LDS_B128 + s_wait_asynccnt (ASYNCcnt).
// ---------------------------------------------------------------------------

typedef float v2f __attribute__((ext_vector_type(2)));
typedef float v8f __attribute__((ext_vector_type(8)));

__device__ __forceinline__ v8f vzero8() {
  v8f c;
#pragma unroll
  for (int r = 0; r < 8; ++r) c[r] = 0.0f;
  return c;
}

// ---------------------------------------------------------------------------
// Cooperative async copy of `rows` rows (clamped at maxrow) x `cols` floats
// from global into LDS via GLOBAL_LOAD_ASYNC_TO_LDS_B128. Caller must
// __syncthreads() afterwards (each wave waits its own ASYNCcnt here).
// Low 32 bits of a generic shared-memory address are the LDS byte offset
// (CDNA5 aperture mapping, ISA 10.2).
// ---------------------------------------------------------------------------
__device__ __forceinline__ void async_copy_rows(float* dst_lds,
                                                const float* __restrict__ src,
                                                int row0, int rows, int cols,
                                                int maxrow) {
  const uint32_t lbase = (uint32_t)(uintptr_t)dst_lds;
  const int f4_per_row = cols >> 2;
  const int total = rows * f4_per_row;
  for (int i = threadIdx.x; i < total; i += blockDim.x) {
    const int r = i / f4_per_row;
    const int cc = i - r * f4_per_row;
    int gr = row0 + r;
    gr = gr < maxrow ? gr : maxrow - 1;
    const uint32_t laddr = lbase + (uint32_t)i * 16u;
    const uint64_t gaddr =
        (uint64_t)(uintptr_t)src + ((uint64_t)gr * cols + (uint64_t)cc * 4) * 4u;
    asm volatile("global_load_async_to_lds_b128 %0, %1, off"
                 :
                 : "v"(laddr), "v"(gaddr)
                 : "memory");
  }
  asm volatile("s_wait_asynccnt 0" ::: "memory");
}

// ---------------------------------------------------------------------------
// Dual-row-tile 16x16 WMMA GEMM: C0 += A0*B, C1 += A1*B, sharing each B
// fragment across two V_WMMA_F32_16X16X4_F32 issues (halves weight loads).
// A0/A1: 16xK row-major (LDS). B: Kx16 row-major (global, L2-resident).
// Fragment layouts per CDNA5 ISA 7.12.2.
// ---------------------------------------------------------------------------
__device__ __forceinline__ void tile_gemm2(const float* __restrict__ A0,
                                           const float* __restrict__ A1,
                                           int lda, const float* __restrict__ B,
                                           int ldb, int K, v8f& c0, v8f& c1) {
  const int lane = threadIdx.x & 31;
  const int half = lane >> 4;  // 0: lanes 0-15, 1: lanes 16-31
  const int m = lane & 15;     // row for A, column for B/C/D
  for (int k = 0; k < K; k += 4) {
    v2f b, a0, a1;
    b.x = B[(k + 2 * half + 0) * ldb + m];
    b.y = B[(k + 2 * half + 1) * ldb + m];
    a0.x = A0[m * lda + k + 2 * half + 0];
    a0.y = A0[m * lda + k + 2 * half + 1];
    a1.x = A1[m * lda + k + 2 * half + 0];
    a1.y = A1[m * lda + k + 2 * half + 1];
    c0 = __builtin_amdgcn_wmma_f32_16x16x4_f32(false, a0, false, b, (short)0,
                                               c0, false, false);
    c1 = __builtin_amdgcn_wmma_f32_16x16x4_f32(false, a1, false, b, (short)0,
                                               c1, false, false);
  }
}

// ---------------------------------------------------------------------------
// h = x @ Wx + bx.  32 rows per block (8 waves); wave w owns columns
// [16w, 16w+16) for both 16-row tiles (B-fragment reuse x2).
// ---------------------------------------------------------------------------
__global__ void __launch_bounds__(256) k_node_embed(
    const float* __restrict__ x, const float* __restrict__ Wx,
    const float* __restrict__ bx, float* __restrict__ h, int N) {
  __shared__ float sx[32 * 128];
  const int row0 = blockIdx.x * 32;
  async_copy_rows(sx, x, row0, 32, 128, N);
  __syncthreads();

  const int wave = threadIdx.x >> 5;
  const int lane = threadIdx.x & 31;
  const int half = lane >> 4, m = lane & 15;
  const int col = wave * 16 + m;

  __builtin_prefetch(Wx + wave * 16, 0, 1);
  v8f c0 = vzero8(), c1 = vzero8();
  tile_gemm2(sx, sx + 16 * 128, 128, Wx + wave * 16, 128, 128, c0, c1);
  const float bias = bx[col];
#pragma unroll
  for (int r = 0; r < 8; ++r) {
    const int ra = row0 + r + 8 * half;
    const int rb = ra + 16;
    if (ra < N) h[ra * 128 + col] = c0[r] + bias;
    if (rb < N) h[rb * 128 + col] = c1[r] + bias;
  }
}

// ---------------------------------------------------------------------------
// Zero agg (n4 float4s) and bn stats (256 floats).
// ---------------------------------------------------------------------------
__global__ void k_zero(float* __restrict__ agg, int n4, float* __restrict__ bn) {
  int i = blockIdx.x * blockDim.x + threadIdx.x;
  float4 z;
  z.x = z.y = z.z = z.w = 0.0f;
  if (i < n4) ((float4*)agg)[i] = z;
  if (i < 64) ((float4*)bn)[i] = z;
}

// ---------------------------------------------------------------------------
// Fused edge pass. 64 edges per block, one wave per edge (8 iterations/wave),
// 4 channels per lane.  We/be async-staged into LDS once per block.
//   eemb = ea[e] @ We_l + be_l   (ea is zero for self-loops)
//   agg[dst] += h[src] + eemb    (f32 atomics, L2-resident h/agg)
// ---------------------------------------------------------------------------
__global__ void __launch_bounds__(256) k_gather_scatter(
    const float* __restrict__ h, const float* __restrict__ ea,
    const int* __restrict__ ei, const float* __restrict__ We_l,
    const float* __restrict__ be_l, float* __restrict__ agg, int E, int Nn) {
  __shared__ float sWe[16 * 128];
  __shared__ float sbe[128];
  async_copy_rows(sWe, We_l, 0, 16, 128, 16);
  async_copy_rows(sbe, be_l, 0, 1, 128, 1);
  __syncthreads();

  const int lane = threadIdx.x & 31;
  const int c0i = lane * 4;
  const int ET = E + Nn;
  const float4 bev = *(const float4*)(sbe + c0i);

  for (int j = threadIdx.x >> 5; j < 64; j += 8) {
    const int e = blockIdx.x * 64 + j;
    if (e >= ET) break;

    float4 acc = bev;
    int src, dst;
    if (e < E) {
      src = ei[e];
      dst = ei[E + e];
      // lanes 0..15 hold the 16 edge features; broadcast via lane shuffle.
      const float av = ea[(long)e * 16 + (lane & 15)];
#pragma unroll
      for (int k = 0; k < 16; ++k) {
        const float ak = __shfl(av, k, 32);
        const float4 w = *(const float4*)(sWe + k * 128 + c0i);
        acc.x += ak * w.x;
        acc.y += ak * w.y;
        acc.z += ak * w.z;
        acc.w += ak * w.w;
      }
    } else {
      src = dst = e - E;  // self loop, zero edge attrs -> eemb == be
    }

    const float4 hv = *(const float4*)(h + (long)src * 128 + c0i);
    acc.x += hv.x;
    acc.y += hv.y;
    acc.z += hv.z;
    acc.w += hv.w;

    float* a = agg + (long)dst * 128 + c0i;
    atomicAdd(a + 0, acc.x);
    atomicAdd(a + 1, acc.y);
    atomicAdd(a + 2, acc.z);
    atomicAdd(a + 3, acc.w);
  }
}

// ---------------------------------------------------------------------------
// Fused GIN MLP per 32-row tile: h2 = relu(agg@W1 + b1)@W2 + b2, plus
// per-channel sum / sum^2 for BatchNorm (atomics into bn[256]).
// LDS: 16KB agg tile + 32KB hidden tile.
// ---------------------------------------------------------------------------
__global__ void __launch_bounds__(256) k_mlp(
    const float* __restrict__ agg, const float* __restrict__ W1_l,
    const float* __restrict__ b1_l, const float* __restrict__ W2_l,
    const float* __restrict__ b2_l, float* __restrict__ h2,
    float* __restrict__ bn, int N) {
  __shared__ float sA[32 * 128];
  __shared__ float sH[32 * 256];
  const int row0 = blockIdx.x * 32;
  async_copy_rows(sA, agg, row0, 32, 128, N);
  __syncthreads();

  const int wave = threadIdx.x >> 5;
  const int lane = threadIdx.x & 31;
  const int half = lane >> 4, m = lane & 15;

  // hidden = relu(agg@W1 + b1): 16 column tiles; wave w does t = w and w+8,
  // each for both 16-row tiles (B reuse).
#pragma unroll
  for (int tt = 0; tt < 2; ++tt) {
    const int t = wave + tt * 8;
    __builtin_prefetch(W1_l + t * 16, 0, 1);
    v8f c0 = vzero8(), c1 = vzero8();
    tile_gemm2(sA, sA + 16 * 128, 128, W1_l + t * 16, 256, 128, c0, c1);
    const float b = b1_l[t * 16 + m];
#pragma unroll
    for (int r = 0; r < 8; ++r) {
      float v0 = c0[r] + b;
      float v1 = c1[r] + b;
      sH[(r + 8 * half) * 256 + t * 16 + m] = v0 > 0.0f ? v0 : 0.0f;
      sH[(16 + r + 8 * half) * 256 + t * 16 + m] = v1 > 0.0f ? v1 : 0.0f;
    }
  }
  __syncthreads();

  // out = hidden@W2 + b2: 8 column tiles, one per wave, K=256, B reuse x2.
  __builtin_prefetch(W2_l + wave * 16, 0, 1);
  v8f c0 = vzero8(), c1 = vzero8();
  tile_gemm2(sH, sH + 16 * 256, 256, W2_l + wave * 16, 128, 256, c0, c1);
  const float b = b2_l[wave * 16 + m];
  const int col = wave * 16 + m;

  float s = 0.0f, sq = 0.0f;
#pragma unroll
  for (int r = 0; r < 8; ++r) {
    const int ra = row0 + r + 8 * half;
    const int rb = ra + 16;
    if (ra < N) {
      const float v = c0[r] + b;
      h2[ra * 128 + col] = v;
      s += v;
      sq += v * v;
    }
    if (rb < N) {
      const float v = c1[r] + b;
      h2[rb * 128 + col] = v;
      s += v;
      sq += v * v;
    }
  }
  // Column `m` lives in lanes m and m+16: combine halves, then one atomic.
  s += __shfl_xor(s, 16, 32);
  sq += __shfl_xor(sq, 16, 32);
  if (lane < 16) {
    atomicAdd(&bn[col], s);
    atomicAdd(&bn[128 + col], sq);
  }
}

// ---------------------------------------------------------------------------
// BatchNorm apply (+ optional relu): out = (h2 - mean)*rsqrt(var+eps)*g + b.
// var = E[x^2] - E[x]^2 (biased), matching the reference.
// ---------------------------------------------------------------------------
__global__ void __launch_bounds__(256) k_bn(
    const float* __restrict__ h2, const float* __restrict__ bn,
    const float* __restrict__ gamma_l, const float* __restrict__ beta_l,
    float* __restrict__ out, int do_relu, int Nn) {
  const int n4 = Nn * 32;  // 128 channels = 32 float4 per row
  const int i = blockIdx.x * blockDim.x + threadIdx.x;
  if (i >= n4) return;
  const int c0 = (i & 31) * 4;
  const float invN = 1.0f / (float)Nn;

  const float4 sm = *(const float4*)(bn + c0);
  const float4 sg = *(const float4*)(bn + 128 + c0);
  const float4 g = *(const float4*)(gamma_l + c0);
  const float4 bt = *(const float4*)(beta_l + c0);
  float4 v = ((const float4*)h2)[i];

  float mean, var, sc, y;
#define BN1(comp)                                     \
  mean = sm.comp * invN;                              \
  var = sg.comp * invN - mean * mean;                 \
  sc = __frsqrt_rn(var + 1e-5f) * g.comp;             \
  y = (v.comp - mean) * sc + bt.comp;                 \
  v.comp = do_relu ? (y > 0.0f ? y : 0.0f) : y;
  BN1(x) BN1(y) BN1(z) BN1(w)
#undef BN1
  ((float4*)out)[i] = v;
}

// ---------------------------------------------------------------------------
extern "C" void kernel_launch(void* const* d_in, const int* in_sizes, int n_in,
                              void* d_out, int out_size, void* d_ws,
                              size_t ws_size, hipStream_t stream) {
  const float* x = (const float*)d_in[0];
  const float* ea = (const float*)d_in[1];
  const int* ei = (const int*)d_in[2];
  const float* Wx = (const float*)d_in[3];
  const float* bx = (const float*)d_in[4];
  const float* We = (const float*)d_in[5];
  const float* be = (const float*)d_in[6];
  const float* W1 = (const float*)d_in[7];
  const float* b1 = (const float*)d_in[8];
  const float* W2 = (const float*)d_in[9];
  const float* b2 = (const float*)d_in[10];
  const float* gam = (const float*)d_in[11];
  const float* bet = (const float*)d_in[12];
  float* out = (float*)d_out;

  const int N = in_sizes[0] / 128;  // 50000
  const int E = in_sizes[1] / 16;   // 800000
  const int L = in_sizes[6] / 128;  // 5
  const int ET = E + N;

  float* h = (float*)d_ws;            // [N,128]
  float* agg = h + (size_t)N * 128;   // [N,128]
  float* h2 = agg + (size_t)N * 128;  // [N,128]
  float* bn = h2 + (size_t)N * 128;   // [256] sum | sumsq

  const int rowTiles32 = (N + 31) / 32;  // 1563
  const int n4 = N * 32;                 // float4s in agg
  const int zeroBlocks = (n4 + 255) / 256;
  const int edgeBlocks = (ET + 63) / 64;  // 64 edges per block
  const int bnBlocks = (n4 + 255) / 256;

  k_node_embed<<<rowTiles32, 256, 0, stream>>>(x, Wx, bx, h, N);

  for (int l = 0; l < L; ++l) {
    const float* We_l = We + (size_t)l * 16 * 128;
    const float* be_l = be + (size_t)l * 128;
    const float* W1_l = W1 + (size_t)l * 128 * 256;
    const float* b1_l = b1 + (size_t)l * 256;
    const float* W2_l = W2 + (size_t)l * 256 * 128;
    const float* b2_l = b2 + (size_t)l * 128;
    const float* g_l = gam + (size_t)l * 128;
    const float* bt_l = bet + (size_t)l * 128;
    float* bn_out = (l == L - 1) ? out : h;

    k_zero<<<zeroBlocks, 256, 0, stream>>>(agg, n4, bn);
    k_gather_scatter<<<edgeBlocks, 256, 0, stream>>>(h, ea, ei, We_l, be_l,
                                                     agg, E, N);
    k_mlp<<<rowTiles32, 256, 0, stream>>>(agg, W1_l, b1_l, W2_l, b2_l, h2, bn,
                                          N);
    k_bn<<<bnBlocks, 256, 0, stream>>>(h2, bn, g_l, bt_l, bn_out,
                                       (l < L - 1) ? 1 : 0, N);
  }
}